// ConvolutionStack_75307956568344
// MI455X (gfx1250) — compile-verified
//
#include <hip/hip_runtime.h>

// WaveNet dilated-conv stack for gfx1250 (MI455X), fp32 via V_WMMA_F32_16X16X4_F32.
// One fused kernel per layer: dilated conv (3 shifted GEMMs) + bias + gated
// activation + skip store + 1x1 output transform + residual, all on WMMA tiles.
// LDS panels are stored channel-PAIR interleaved so every WMMA B-fragment is a
// single aligned ds_load_b64 into an even VGPR pair (no v_mov patch-up).

typedef __attribute__((ext_vector_type(2))) float v2f;
typedef __attribute__((ext_vector_type(8))) float v8f;

#define CH      64
#define TWO_CH  128
#define TLEN    16384
#define NB      8
#define NLAYERS 10
#define KW      3
#define TILE_T  128
#define SUB_T   16
#define NSUB    (TILE_T / SUB_T)
#define PAIRS   (CH / 2)              // 32 channel pairs

__device__ __forceinline__ float sigmoid_f(float z) {
  return 1.0f / (1.0f + __expf(-z));
}
__device__ __forceinline__ float tanh_f(float z) {
  float zc = fminf(fmaxf(z, -15.0f), 15.0f);
  float e = __expf(-2.0f * zc);
  return (1.0f - e) / (1.0f + e);
}

__launch_bounds__(256)
__global__ void wavenet_layer_kernel(
    const float* __restrict__ xin,   // (B, CH, T)   layer input
    float* __restrict__ xout,        // (B, CH, T)   layer output (residual path)
    float* __restrict__ skip,        // (B, CH, T)   skip output (= gated act)
    const float* __restrict__ wconv, // (2CH, CH, 3)
    const float* __restrict__ bconv, // (2CH)
    const float* __restrict__ wout,  // (CH, CH)
    const float* __restrict__ bout,  // (CH)
    int dil, int has_out)
{
  // pair-interleaved panels: element [p*SUB_T + t] = {ch 2p, ch 2p+1} at time t
  __shared__ v2f xs2[KW * PAIRS * SUB_T];  // 12KB: 3 dilation-shifted panels
  __shared__ v2f ys2[PAIRS * SUB_T];       //  4KB: sigmoid-half pre-activations
  __shared__ v2f a2[PAIRS * SUB_T];        //  4KB: gated activations (1x1 B-panel)

  const int tid  = threadIdx.x;
  const int lane = tid & 31;
  const int wave = tid >> 5;      // 0..7, each owns a 16-row conv output tile
  const int lm   = lane & 15;
  const int half = lane >> 4;     // 0 or 1 (WMMA half-wave)

  const int  b     = blockIdx.y;
  const int  t0    = blockIdx.x * TILE_T;
  const long xbase = (long)b * CH * TLEN;

  // ---- A-fragments of the dilated-conv weights (registers, per wave) ----
  // A-matrix 16x4 layout: lane l holds row m=l%16, K = 2*(l/16) + {0,1}.
  const int mrow = 16 * wave + lm;
  v2f aW[KW * 16];
#pragma unroll
  for (int k = 0; k < KW; ++k) {
#pragma unroll
    for (int kc = 0; kc < 16; ++kc) {
      int i0 = 4 * kc + 2 * half;
      v2f f;
      f.x = wconv[(mrow * CH + i0) * KW + k];
      f.y = wconv[(mrow * CH + i0 + 1) * KW + k];
      aW[k * 16 + kc] = f;
    }
  }
  float bc[8];
#pragma unroll
  for (int r = 0; r < 8; ++r)
    bc[r] = bconv[16 * wave + r + 8 * half];

  // waves 0..3 additionally own 16 rows of the 1x1 output transform
  v2f   aO[16];
  float bo[8];
  if (wave < 4) {
#pragma unroll
    for (int kc = 0; kc < 16; ++kc) {
      int i0 = 4 * kc + 2 * half;
      aO[kc] = *(const v2f*)&wout[(16 * wave + lm) * CH + i0];  // consecutive -> b64
    }
#pragma unroll
    for (int r = 0; r < 8; ++r)
      bo[r] = bout[16 * wave + r + 8 * half];
  }

  for (int s = 0; s < NSUB; ++s) {
    const int ts = t0 + s * SUB_T;

    // ---- stage the 3 dilation-shifted panels (pair-interleaved) into LDS ----
    for (int idx = tid; idx < KW * PAIRS * SUB_T; idx += 256) {   // 6 iters
      int tap = idx / (PAIRS * SUB_T);
      int rem = idx % (PAIRS * SUB_T);
      int p   = rem >> 4;
      int tt  = rem & 15;
      int t   = ts - (2 - tap) * dil + tt;   // causal left pad -> zeros
      v2f v = {0.0f, 0.0f};
      if (t >= 0) {
        v.x = xin[xbase + (long)(2 * p) * TLEN + t];
        v.y = xin[xbase + (long)(2 * p + 1) * TLEN + t];
      }
      xs2[idx] = v;
    }
    __syncthreads();

    // ---- dilated conv: D(16x16) += A(16x4) x B(4x16), 3 taps x 16 K-chunks ----
    v8f acc = {};
#pragma unroll
    for (int k = 0; k < KW; ++k) {
#pragma unroll
      for (int kc = 0; kc < 16; ++kc) {
        v2f bf = xs2[k * (PAIRS * SUB_T) + (2 * kc + half) * SUB_T + lm];
        acc = __builtin_amdgcn_wmma_f32_16x16x4_f32(
            false, aW[k * 16 + kc], false, bf, (short)0, acc, false, false);
      }
    }

    // ---- gated activation: rows 0..63 tanh-half, rows 64..127 sigmoid-half ----
    if (wave >= 4) {
#pragma unroll
      for (int r = 0; r < 8; r += 2) {
        int grow = 16 * (wave - 4) + r + 8 * half;   // gate row (0..63), even
        v2f v;
        v.x = acc[r] + bc[r];
        v.y = acc[r + 1] + bc[r + 1];
        ys2[(grow >> 1) * SUB_T + lm] = v;
      }
    }
    __syncthreads();

    if (wave < 4) {
#pragma unroll
      for (int r = 0; r < 8; r += 2) {
        int row = 16 * wave + r + 8 * half;          // even row pair (row, row+1)
        int pr  = row >> 1;
        v2f ysv = ys2[pr * SUB_T + lm];
        float a0 = tanh_f(acc[r] + bc[r]) * sigmoid_f(ysv.x);
        float a1 = tanh_f(acc[r + 1] + bc[r + 1]) * sigmoid_f(ysv.y);
        skip[xbase + (long)row * TLEN + ts + lm]       = a0;
        skip[xbase + (long)(row + 1) * TLEN + ts + lm] = a1;
        v2f av; av.x = a0; av.y = a1;
        a2[pr * SUB_T + lm] = av;
        if (!has_out) {   // last layer: no output transform, x = a + residual
          v2f hv = xs2[2 * (PAIRS * SUB_T) + pr * SUB_T + lm];  // tap2 = unshifted
          xout[xbase + (long)row * TLEN + ts + lm]       = a0 + hv.x;
          xout[xbase + (long)(row + 1) * TLEN + ts + lm] = a1 + hv.y;
        }
      }
    }
    __syncthreads();

    // ---- 1x1 output transform + residual (waves 0..3) ----
    if (has_out && wave < 4) {
      v8f acc2 = {};
#pragma unroll
      for (int kc = 0; kc < 16; ++kc) {
        v2f bf = a2[(2 * kc + half) * SUB_T + lm];
        acc2 = __builtin_amdgcn_wmma_f32_16x16x4_f32(
            false, aO[kc], false, bf, (short)0, acc2, false, false);
      }
#pragma unroll
      for (int r = 0; r < 8; r += 2) {
        int row = 16 * wave + r + 8 * half;
        int pr  = row >> 1;
        v2f hv = xs2[2 * (PAIRS * SUB_T) + pr * SUB_T + lm];
        xout[xbase + (long)row * TLEN + ts + lm]       = acc2[r] + bo[r] + hv.x;
        xout[xbase + (long)(row + 1) * TLEN + ts + lm] = acc2[r + 1] + bo[r + 1] + hv.y;
      }
    }
    __syncthreads();  // protect xs2/a2 before next subtile's staging
  }
}

extern "C" void kernel_launch(void* const* d_in, const int* in_sizes, int n_in,
                              void* d_out, int out_size, void* d_ws, size_t ws_size,
                              hipStream_t stream) {
  (void)in_sizes; (void)n_in; (void)out_size; (void)ws_size;

  const float* x     = (const float*)d_in[0];
  const float* wconv = (const float*)d_in[1];
  const float* bconv = (const float*)d_in[2];
  const float* wout  = (const float*)d_in[3];
  const float* bout  = (const float*)d_in[4];
  float* out = (float*)d_out;

  const long plane = (long)NB * CH * TLEN;   // 8*64*16384 elements (32 MB)
  float* pA = out;            // final-x slot of d_out doubles as ping buffer
  float* pB = (float*)d_ws;   // 32 MB scratch = pong buffer

  static const int dil[NLAYERS] = {1, 2, 4, 8, 16, 32, 64, 128, 256, 512};

  dim3 grid(TLEN / TILE_T, NB);   // 128 x 8 blocks
  for (int i = 0; i < NLAYERS; ++i) {
    // parity chosen so layer 9's x lands exactly in d_out[0]
    float*       xo = (i & 1) ? pA : pB;
    const float* xi = (i == 0) ? x : ((i & 1) ? pB : pA);
    float*       sk = out + (long)(1 + i) * plane;
    wavenet_layer_kernel<<<grid, 256, 0, stream>>>(
        xi, xo, sk,
        wconv + (long)i * TWO_CH * CH * KW,
        bconv + (long)i * TWO_CH,
        wout  + (long)i * CH * CH,
        bout  + (long)i * CH,
        dil[i], (i < NLAYERS - 1) ? 1 : 0);
  }
}